// MostPopularEncoder_21148418965663
// MI455X (gfx1250) — compile-verified
//
#include <hip/hip_runtime.h>
#include <hip/hip_bf16.h>

// Problem constants (match reference)
#define B_       32
#define L_       4096
#define C_       128
#define OUTW     257          // 1 (delta) + 128 (probs) + 128 (amt)
#define MOM      0.9f
#define OMM      0.1f         // 1 - momentum
#define MAXDT    100.0f

// Parallel-scan decomposition: chunks of 256 outputs, 256-step warm-up.
// 0.9^256 ~ 2e-12 << fp32 eps relative to O(0.1) carries, so zero-init
// warm-up reproduces the exact sequential scan to fp32 precision.
#define CHUNK    256
#define LOOKBACK 256
#define NCHUNK   (L_ / CHUNK)  // 16

typedef __attribute__((ext_vector_type(2))) float v2f;
typedef __attribute__((ext_vector_type(8))) float v8f;

// 0.9^e for e in [0,16], via repeated multiply (matches fp32 rounding chain)
__device__ __forceinline__ float mpow(int e) {
    float r = 1.0f;
#pragma unroll
    for (int i = 0; i < 16; ++i) r = (i < e) ? r * MOM : r;
    return r;
}

// One workgroup = (batch row b, output chunk k). 8 waves; wave w owns
// channels [16w, 16w+16). Each 16-step block is evaluated as
//   D[16x16] = W[16x16] * U[16x16]  +  decay[16] (x) carry_in[16]
// using 5 V_WMMA_F32_16X16X4_F32 per state (probs, amt).
__global__ __launch_bounds__(256) void ema_wmma_kernel(
    const int*   __restrict__ labels,
    const float* __restrict__ amounts,
    float*       __restrict__ out)
{
    const int b     = blockIdx.y;
    const int chunk = blockIdx.x;
    const int lane  = threadIdx.x & 31;
    const int wave  = threadIdx.x >> 5;      // channel tile 0..7
    const int n     = lane & 15;             // N (column) within tile
    const bool hi   = lane >= 16;
    const int cb    = wave * 16;             // first channel of this tile
    const int ch    = cb + n;                // this lane's channel

    const int*   lab  = labels  + (size_t)b * L_;
    const float* am   = amounts + (size_t)b * L_;
    float*       orow = out + (size_t)b * L_ * OUTW;

    // ---- constant A operands (triangular decay matrix W, 4 K-slices) ----
    // A 16x4 fp32 layout: lanes 0-15 & 16-31 both hold M = lane&15;
    // VGPR0 = cols {K0 | K2}, VGPR1 = cols {K1 | K3}.
    const int M = n;                          // output row within block
    v2f A[4];
#pragma unroll
    for (int s = 0; s < 4; ++s) {
        const int j0 = 4 * s + (hi ? 2 : 0);
        const int j1 = 4 * s + (hi ? 3 : 1);
        A[s].x = (M >= j0) ? mpow(M - j0) : 0.0f;
        A[s].y = (M >= j1) ? mpow(M - j1) : 0.0f;
    }
    // Carry-fold column: A_dec[l,0] = m^(l+1), other cols 0.
    v2f Adec;
    Adec.x = hi ? 0.0f : mpow(M + 1);
    Adec.y = 0.0f;

    const int chunkStart = chunk * CHUNK;
    const int start = (chunkStart >= LOOKBACK) ? (chunkStart - LOOKBACK) : 0;
    const int end   = chunkStart + CHUNK;

    // Running carry for channel cb+n (meaningful in lanes 0..15).
    float carryP = 0.0f;
    float carryA = 0.0f;

    for (int t = start; t < end; t += 16) {
        if (t + 16 < end) {
            __builtin_prefetch(lab + t + 16, 0, 0);   // global_prefetch_b8
            __builtin_prefetch(am  + t + 16, 0, 0);
        }

        // --- fold carry_in:  acc = Adec * Bcarry  (Bcarry row K0 = carry) ---
        // B 4x16 fp32 layout: VGPR0 = rows {K0 | K2}, VGPR1 = rows {K1 | K3}.
        v2f bcP; bcP.x = hi ? 0.0f : carryP; bcP.y = 0.0f;
        v2f bcA; bcA.x = hi ? 0.0f : carryA; bcA.y = 0.0f;
        v8f zero = {};
        v8f accP = __builtin_amdgcn_wmma_f32_16x16x4_f32(
            false, Adec, false, bcP, (short)0, zero, false, false);
        v8f accA = __builtin_amdgcn_wmma_f32_16x16x4_f32(
            false, Adec, false, bcA, (short)0, zero, false, false);

        // --- 4 K-slices of the triangular block matmul ---
#pragma unroll
        for (int s = 0; s < 4; ++s) {
            const int base = t + 4 * s;
            // Uniform (wave-scalar) loads of this slice's labels/amounts.
            const int   l0 = lab[base + 0], l1 = lab[base + 1];
            const int   l2 = lab[base + 2], l3 = lab[base + 3];
            const float a0 = am[base + 0],  a1 = am[base + 1];
            const float a2 = am[base + 2],  a3 = am[base + 3];
            // EMA scan seeds with x[0] itself (weight 1), else weight 0.1.
            const float w0 = (base == 0) ? 1.0f : OMM;

            // Sparse one-hot rows of U for this lane's channel.
            const float p0 = (l0 == ch) ? w0  : 0.0f;
            const float p1 = (l1 == ch) ? OMM : 0.0f;
            const float p2 = (l2 == ch) ? OMM : 0.0f;
            const float p3 = (l3 == ch) ? OMM : 0.0f;

            v2f bp, ba;
            bp.x = hi ? p2        : p0;
            bp.y = hi ? p3        : p1;
            ba.x = hi ? p2 * a2   : p0 * a0;
            ba.y = hi ? p3 * a3   : p1 * a1;

            accP = __builtin_amdgcn_wmma_f32_16x16x4_f32(
                false, A[s], false, bp, (short)0, accP, false, false);
            accA = __builtin_amdgcn_wmma_f32_16x16x4_f32(
                false, A[s], false, ba, (short)0, accA, false, false);
        }

        // --- carry_out = D row 15 = VGPR7 of lanes 16..31 (C/D layout) ---
        carryP = __shfl(accP[7], 16 + n, 32);
        carryA = __shfl(accA[7], 16 + n, 32);

        // --- store outputs (only inside the owned chunk) ---
        if (t >= chunkStart) {
#pragma unroll
            for (int r = 0; r < 8; ++r) {
                const int l = t + r + (hi ? 8 : 0);   // M = r (lo) / r+8 (hi)
                const float p = accP[r];
                const float a = accA[r] / fmaxf(p, 1e-6f);
                const size_t o = (size_t)l * OUTW;
                orow[o + 1   + ch] = p;   // probs (row-sum is exactly 1)
                orow[o + 129 + ch] = a;   // amt
            }
        }
    }
}

// Scalar delta-EMA channel: one thread per (row, chunk), same warm-up trick.
// x[0] = 0 for deltas, so the uniform recurrence c = m*c + 0.1*d is exact
// at l == 0 as well.
__global__ __launch_bounds__(256) void delta_kernel(
    const float* __restrict__ ts,
    float*       __restrict__ out)
{
    const int tid = blockIdx.x * blockDim.x + threadIdx.x;
    if (tid >= B_ * NCHUNK) return;
    const int b = tid / NCHUNK;
    const int k = tid % NCHUNK;
    const int chunkStart = k * CHUNK;
    const int start = (chunkStart >= LOOKBACK) ? (chunkStart - LOOKBACK) : 0;
    const float* tr   = ts + (size_t)b * L_;
    float*       orow = out + (size_t)b * L_ * OUTW;

    float prev  = (start > 0) ? tr[start - 1] : 0.0f;
    float carry = 0.0f;
    for (int l = start; l < chunkStart + CHUNK; ++l) {
        const float cur = tr[l];
        const float d = (l == 0) ? 0.0f : fminf(cur - prev, MAXDT);
        prev = cur;
        carry = MOM * carry + OMM * d;
        if (l >= chunkStart) orow[(size_t)l * OUTW] = carry;
    }
}

extern "C" void kernel_launch(void* const* d_in, const int* in_sizes, int n_in,
                              void* d_out, int out_size, void* d_ws, size_t ws_size,
                              hipStream_t stream) {
    const float* timestamps = (const float*)d_in[0];
    const int*   labels     = (const int*)d_in[1];
    const float* amounts    = (const float*)d_in[2];
    float* out = (float*)d_out;

    dim3 grid(NCHUNK, B_);            // 16 x 32 = 512 workgroups
    ema_wmma_kernel<<<grid, 256, 0, stream>>>(labels, amounts, out);

    const int nd = B_ * NCHUNK;       // 512 threads
    delta_kernel<<<(nd + 255) / 256, 256, 0, stream>>>(timestamps, out);
}